// T2I_OT_AdapGating_Fusion_8340826489176
// MI455X (gfx1250) — compile-verified
//
#include <hip/hip_runtime.h>
#include <hip/hip_bf16.h>

// ---------------------------------------------------------------------------
// T2I optimal-transport adaptive gating fusion, CDNA5 (gfx1250), wave32 WMMA.
// B=2, C=64, N=256, D=16, H=64, W=64  ->  L=B*C=128, M=D*H*W=65536
// ---------------------------------------------------------------------------

#define L_DIM 128
#define N_DIM 256
#define M_DIM 65536
#define C_DIM 64
#define B_DIM 2
#define W_DIM 64
#define ACHUNKS 64  // split-K chunks for the aligned-text GEMM

typedef float v2f __attribute__((ext_vector_type(2)));
typedef float v8f __attribute__((ext_vector_type(8)));

static __device__ __forceinline__ v8f wmma_f32(v2f a, v2f b, v8f c) {
  // V_WMMA_F32_16X16X4_F32: (neg_a, A, neg_b, B, c_mod, C, reuse_a, reuse_b)
  return __builtin_amdgcn_wmma_f32_16x16x4_f32(false, a, false, b, (short)0, c,
                                               false, false);
}

// 8-byte vector load of two contiguous floats (caller guarantees alignment).
static __device__ __forceinline__ v2f ldg2(const float* __restrict__ p) {
  return *(const v2f*)p;
}

// ---------------------------------------------------------------------------
// 1) Column squared norms: tn[N] over text columns, im[M] over image columns.
// ---------------------------------------------------------------------------
__global__ void k_col_norms(const float* __restrict__ text,
                            const float* __restrict__ image,
                            float* __restrict__ tn, float* __restrict__ im) {
  int tid = blockIdx.x * blockDim.x + threadIdx.x;
  if (tid < N_DIM) {
    float s = 0.0f;
    for (int l = 0; l < L_DIM; ++l) {
      float x = text[l * N_DIM + tid];
      s += x * x;
    }
    tn[tid] = s;
  }
  if (tid < M_DIM) {
    float s = 0.0f;
    for (int l = 0; l < L_DIM; ++l) {
      float x = image[(size_t)l * M_DIM + tid];
      s += x * x;
    }
    im[tid] = s;
  }
}

// ---------------------------------------------------------------------------
// 2) exp_term[i,j] = exp(-sqrt(max(tn[i]+im[j]-2*cross[i,j], 1e-12)))
//    cross = tf^T @ imf via V_WMMA_F32_16X16X4_F32, one 16x16 tile per wave.
// ---------------------------------------------------------------------------
__global__ void __launch_bounds__(256) k_exp_term(
    const float* __restrict__ text, const float* __restrict__ image,
    const float* __restrict__ tn, const float* __restrict__ im,
    float* __restrict__ e) {
  int wave = threadIdx.x >> 5;
  int lane = threadIdx.x & 31;
  int wid = blockIdx.x * (blockDim.x >> 5) + wave;  // 0..65535
  int n0 = (wid & 15) * 16;                         // 16 n-tiles
  int m0 = (wid >> 4) * 16;                         // 4096 m-tiles
  int col = lane & 15;                              // A-row / B-col index
  int khalf = (lane >> 4) << 1;                     // 0 or 2 (K split)

  v8f acc = {};
  for (int k0 = 0; k0 < L_DIM; k0 += 4) {
    v2f a, b;
    a.x = text[(k0 + khalf) * N_DIM + n0 + col];
    a.y = text[(k0 + khalf + 1) * N_DIM + n0 + col];
    b.x = image[(size_t)(k0 + khalf) * M_DIM + m0 + col];
    b.y = image[(size_t)(k0 + khalf + 1) * M_DIM + m0 + col];
    acc = wmma_f32(a, b, acc);
  }
  int rbase = n0 + ((lane >> 4) << 3);  // C/D: lanes16-31 hold rows M+8
  float imj = im[m0 + col];
#pragma unroll
  for (int g = 0; g < 8; ++g) {
    int i = rbase + g;
    float sq = tn[i] + imj - 2.0f * acc[g];
    float cost = sqrtf(fmaxf(sq, 1e-12f));
    e[(size_t)i * M_DIM + m0 + col] = expf(-cost);  // EPSILON == 1
  }
}

// ---------------------------------------------------------------------------
// 3) Sinkhorn iterations (u[256], v[65536]) with convergence freeze.
//    scal[0]=conv flag; du2arr[256], dv2arr[256] = deterministic diff^2 parts
// ---------------------------------------------------------------------------
__global__ void k_sink_init(float* __restrict__ u, float* __restrict__ v,
                            float* __restrict__ scal) {
  int tid = blockIdx.x * blockDim.x + threadIdx.x;
  if (tid < M_DIM) v[tid] = 1.0f;
  if (tid < N_DIM) u[tid] = 1.0f;
  if (tid < 4) scal[tid] = 0.0f;
}

__global__ void k_sink_u(const float* __restrict__ e, float* __restrict__ u,
                         const float* __restrict__ v,
                         const float* __restrict__ scal,
                         float* __restrict__ du2arr) {
  if (scal[0] != 0.0f) return;  // converged: u frozen (uniform branch)
  __shared__ float red[256];
  int i = blockIdx.x;
  const float4* e4 = (const float4*)(e + (size_t)i * M_DIM);
  const float4* v4 = (const float4*)v;
  float s = 0.0f;
  for (int j = threadIdx.x; j < M_DIM / 4; j += 256) {
    float4 a = e4[j];
    float4 b = v4[j];
    s += a.x * b.x + a.y * b.y + a.z * b.z + a.w * b.w;
  }
  red[threadIdx.x] = s;
  __syncthreads();
  for (int off = 128; off > 0; off >>= 1) {
    if (threadIdx.x < off) red[threadIdx.x] += red[threadIdx.x + off];
    __syncthreads();
  }
  if (threadIdx.x == 0) {
    float un = 1.0f / (red[0] + 100.0f);
    float d = un - u[i];
    du2arr[i] = d * d;
    u[i] = un;
  }
}

__global__ void k_sink_v(const float* __restrict__ e,
                         const float* __restrict__ u, float* __restrict__ v,
                         const float* __restrict__ scal,
                         float* __restrict__ dv2arr) {
  if (scal[0] != 0.0f) return;  // uniform
  __shared__ float red[256];
  int j = blockIdx.x * 256 + threadIdx.x;
  float s = 0.0f;
  for (int i = 0; i < N_DIM; ++i) s += e[(size_t)i * M_DIM + j] * u[i];
  float vn = 1.0f / (s + 100.0f);
  float d = vn - v[j];
  v[j] = vn;
  red[threadIdx.x] = d * d;
  __syncthreads();
  for (int off = 128; off > 0; off >>= 1) {
    if (threadIdx.x < off) red[threadIdx.x] += red[threadIdx.x + off];
    __syncthreads();
  }
  if (threadIdx.x == 0) dv2arr[blockIdx.x] = red[0];
}

__global__ void k_sink_conv(float* __restrict__ scal,
                            const float* __restrict__ du2arr,
                            const float* __restrict__ dv2arr) {
  if (scal[0] != 0.0f) return;  // uniform
  __shared__ float red[256];
  __shared__ float du2s;
  red[threadIdx.x] = du2arr[threadIdx.x];
  __syncthreads();
  for (int off = 128; off > 0; off >>= 1) {
    if (threadIdx.x < off) red[threadIdx.x] += red[threadIdx.x + off];
    __syncthreads();
  }
  if (threadIdx.x == 0) du2s = red[0];
  __syncthreads();
  red[threadIdx.x] = dv2arr[threadIdx.x];
  __syncthreads();
  for (int off = 128; off > 0; off >>= 1) {
    if (threadIdx.x < off) red[threadIdx.x] += red[threadIdx.x + off];
    __syncthreads();
  }
  if (threadIdx.x == 0) {
    // norm(du)<0.01 && norm(dv)<0.01  <=>  du2<1e-4 && dv2<1e-4
    if (du2s < 1e-4f && red[0] < 1e-4f) scal[0] = 1.0f;
  }
}

// ---------------------------------------------------------------------------
// 4) aligned[l,i] = u[i] * sum_j imf[l,j]*v[j]*e[i,j]   (WMMA, K=65536)
//    Split-K over ACHUNKS chunks -> deterministic partials + reduce.
//    K is the contiguous dim for both operands -> 8B vector loads per lane.
// ---------------------------------------------------------------------------
__global__ void __launch_bounds__(256) k_aligned_partial(
    const float* __restrict__ image, const float* __restrict__ e,
    const float* __restrict__ v, float* __restrict__ partial) {
  int wave = threadIdx.x >> 5;
  int lane = threadIdx.x & 31;
  int wid = blockIdx.x * (blockDim.x >> 5) + wave;  // 0..8191
  int chunk = wid >> 7;                             // 64 K-chunks
  int tile = wid & 127;                             // 128 output tiles
  int l0 = (tile & 7) * 16;                         // 8 l-tiles
  int i0 = (tile >> 3) * 16;                        // 16 i-tiles
  int col = lane & 15;
  int khalf = (lane >> 4) << 1;
  int jbase = chunk * (M_DIM / ACHUNKS);
  int jend = jbase + (M_DIM / ACHUNKS);

  const float* arow = image + (size_t)(l0 + col) * M_DIM;
  const float* brow = e + (size_t)(i0 + col) * M_DIM;

  v8f acc = {};
  for (int j0 = jbase; j0 < jend; j0 += 4) {
    v2f vv = ldg2(v + j0 + khalf);       // global_load_b64
    v2f a = ldg2(arow + j0 + khalf);     // global_load_b64
    v2f b = ldg2(brow + j0 + khalf);     // global_load_b64
    a = a * vv;                          // fold v[j] into A operand
    acc = wmma_f32(a, b, acc);
  }
  int rbase = l0 + ((lane >> 4) << 3);
  float* pdst = partial + (size_t)chunk * (L_DIM * N_DIM);
#pragma unroll
  for (int g = 0; g < 8; ++g)
    pdst[(rbase + g) * N_DIM + i0 + col] = acc[g];
}

__global__ void k_aligned_reduce(const float* __restrict__ partial,
                                 const float* __restrict__ u,
                                 float* __restrict__ aligned) {
  int tid = blockIdx.x * blockDim.x + threadIdx.x;  // 0..32767
  if (tid >= L_DIM * N_DIM) return;
  float s = 0.0f;
  for (int c = 0; c < ACHUNKS; ++c)
    s += partial[(size_t)c * (L_DIM * N_DIM) + tid];
  aligned[tid] = s * u[tid & (N_DIM - 1)];
}

// ---------------------------------------------------------------------------
// 5) Trilinear interp along last axis only: line[b,c,w], w=0..63 from N=256.
// ---------------------------------------------------------------------------
__global__ void k_line(const float* __restrict__ aligned,
                       float* __restrict__ line) {
  int tid = blockIdx.x * blockDim.x + threadIdx.x;
  if (tid >= B_DIM * C_DIM * W_DIM) return;
  int w = tid & 63;
  int row = tid >> 6;  // b*64 + c
  float src = fmaxf((w + 0.5f) * ((float)N_DIM / (float)W_DIM) - 0.5f, 0.0f);
  int x0 = (int)floorf(src);
  if (x0 > N_DIM - 1) x0 = N_DIM - 1;
  int x1 = x0 + 1;
  if (x1 > N_DIM - 1) x1 = N_DIM - 1;
  float lam = src - (float)x0;
  line[tid] =
      aligned[row * N_DIM + x0] * (1.0f - lam) + aligned[row * N_DIM + x1] * lam;
}

// ---------------------------------------------------------------------------
// 6) Fused gating: hid = relu(W1·[t_feat;image]+b1); gate = sigm(W2·hid+b2);
//    out = image*(1-gate)+t_feat*gate.  Per wave: one 16-voxel tile, all 64
//    output channels via 4 WMMA accumulators; hid staged through LDS to
//    re-shape accumulator layout into the B-operand layout for conv2.
// ---------------------------------------------------------------------------
__global__ void __launch_bounds__(256) k_gate(
    const float* __restrict__ image, const float* __restrict__ line,
    const float* __restrict__ W1, const float* __restrict__ b1,
    const float* __restrict__ W2, const float* __restrict__ b2,
    float* __restrict__ out) {
  __shared__ float lhid[8][C_DIM * 16];  // per-wave 64ch x 16vox, 32KB

  int wave = threadIdx.x >> 5;
  int lane = threadIdx.x & 31;
  int gt = blockIdx.x * 8 + wave;  // 0..8191 global voxel-tile id
  int b = gt >> 12;                // / 4096
  int m0 = (gt & 4095) * 16;
  int col = lane & 15;
  int khalf = (lane >> 4) << 1;
  int voxel = m0 + col;
  int wmod = voxel & 63;

  // ---- conv1: 64x128 @ 128x16, K-loop in steps of 4 ----
  v8f acc[4] = {};
  for (int k0 = 0; k0 < 2 * C_DIM; k0 += 4) {
    int i0 = k0 + khalf, i1 = i0 + 1;
    v2f bb;
    bb.x = (i0 < C_DIM) ? line[(b * C_DIM + i0) * W_DIM + wmod]
                        : image[(size_t)(b * C_DIM + i0 - C_DIM) * M_DIM + voxel];
    bb.y = (i1 < C_DIM) ? line[(b * C_DIM + i1) * W_DIM + wmod]
                        : image[(size_t)(b * C_DIM + i1 - C_DIM) * M_DIM + voxel];
#pragma unroll
    for (int t = 0; t < 4; ++t) {
      v2f aa = ldg2(W1 + (t * 16 + col) * (2 * C_DIM) + i0);  // b64
      acc[t] = wmma_f32(aa, bb, acc[t]);
    }
  }

  // bias + relu, stage hid into LDS in B-operand-friendly [och][vox] layout
  int ochbase = (lane >> 4) << 3;
#pragma unroll
  for (int t = 0; t < 4; ++t) {
#pragma unroll
    for (int g = 0; g < 8; ++g) {
      int och = t * 16 + ochbase + g;
      lhid[wave][och * 16 + col] = fmaxf(acc[t][g] + b1[och], 0.0f);
    }
  }
  __syncthreads();

  // ---- conv2: 64x64 @ 64x16, then sigmoid gate + final blend ----
#pragma unroll
  for (int t2 = 0; t2 < 4; ++t2) {
    v8f a2 = {};
    for (int k0 = 0; k0 < C_DIM; k0 += 4) {
      v2f aa = ldg2(W2 + (t2 * 16 + col) * C_DIM + k0 + khalf);  // b64
      v2f bb;
      bb.x = lhid[wave][(k0 + khalf) * 16 + col];
      bb.y = lhid[wave][(k0 + khalf + 1) * 16 + col];
      a2 = wmma_f32(aa, bb, a2);
    }
#pragma unroll
    for (int g = 0; g < 8; ++g) {
      int och = t2 * 16 + ochbase + g;
      float gate = 1.0f / (1.0f + expf(-(a2[g] + b2[och])));
      size_t oidx = (size_t)(b * C_DIM + och) * M_DIM + voxel;
      float img = image[oidx];
      float tfv = line[(b * C_DIM + och) * W_DIM + wmod];
      out[oidx] = img * (1.0f - gate) + tfv * gate;
    }
  }
}

// ---------------------------------------------------------------------------
// Host launcher
// ---------------------------------------------------------------------------
extern "C" void kernel_launch(void* const* d_in, const int* in_sizes, int n_in,
                              void* d_out, int out_size, void* d_ws,
                              size_t ws_size, hipStream_t stream) {
  const float* text = (const float*)d_in[0];   // [B,C,N]
  const float* image = (const float*)d_in[1];  // [B,C,D,H,W]
  const float* W1 = (const float*)d_in[2];     // [C, 2C]
  const float* b1 = (const float*)d_in[3];     // [C]
  const float* W2 = (const float*)d_in[4];     // [C, C]
  const float* b2 = (const float*)d_in[5];     // [C]
  float* out = (float*)d_out;                  // [B,C,D,H,W]

  // Workspace carve-up (fp32), ~76 MB total.
  float* wsf = (float*)d_ws;
  float* e = wsf;                                 // 256*65536
  float* tn = e + (size_t)N_DIM * M_DIM;          // 256
  float* im = tn + N_DIM;                         // 65536
  float* u = im + M_DIM;                          // 256
  float* v = u + N_DIM;                           // 65536
  float* scal = v + M_DIM;                        // 4: conv flag + pad
  float* du2arr = scal + 4;                       // 256
  float* dv2arr = du2arr + N_DIM;                 // 256
  float* partial = dv2arr + N_DIM;                // 64 * 128*256
  float* aligned = partial + (size_t)ACHUNKS * L_DIM * N_DIM;  // 128*256
  float* line = aligned + (size_t)L_DIM * N_DIM;  // 2*64*64

  // 1) column norms
  k_col_norms<<<M_DIM / 256, 256, 0, stream>>>(text, image, tn, im);
  // 2) exp(-cost) matrix: 65536 16x16 WMMA tiles, 8 waves/block
  k_exp_term<<<8192, 256, 0, stream>>>(text, image, tn, im, e);
  // 3) Sinkhorn
  k_sink_init<<<M_DIM / 256, 256, 0, stream>>>(u, v, scal);
  for (int it = 0; it < 20; ++it) {
    k_sink_u<<<N_DIM, 256, 0, stream>>>(e, u, v, scal, du2arr);
    k_sink_v<<<M_DIM / 256, 256, 0, stream>>>(e, u, v, scal, dv2arr);
    k_sink_conv<<<1, 256, 0, stream>>>(scal, du2arr, dv2arr);
  }
  // 4) aligned text: split-K WMMA partials (8192 waves) + deterministic reduce
  k_aligned_partial<<<1024, 256, 0, stream>>>(image, e, v, partial);
  k_aligned_reduce<<<(L_DIM * N_DIM) / 256, 256, 0, stream>>>(partial, u,
                                                              aligned);
  // 5) 1D interpolation line
  k_line<<<(B_DIM * C_DIM * W_DIM) / 256, 256, 0, stream>>>(aligned, line);
  // 6) fused conv-gate-blend
  k_gate<<<1024, 256, 0, stream>>>(image, line, W1, b1, W2, b2, out);
}